// SoftMaskingModule_25271587570014
// MI455X (gfx1250) — compile-verified
//
#include <hip/hip_runtime.h>
#include <cstdint>

// Problem constants (from reference)
#define B_      2
#define S_      4096
#define V_      50257
#define H_      1024
#define MASK_ID_ 50256

#define THREADS 256
#define CHUNK   4096                 // floats per pipeline chunk stride; only THREADS * 4 = 1024 are loaded per chunk
#define NFULL   (V_ / CHUNK)         // 12 full chunks (49152 floats)
#define TAIL_START (NFULL * CHUNK)   // 49152; tail = 1105 floats, handled with guarded loads

// ---- CDNA5 async global->LDS (inline asm: portable across ROCm7.2 / amdgpu-toolchain) ----
__device__ __forceinline__ void async_load_f4_to_lds(uint32_t lds_byte_off,
                                                     uint32_t glb_byte_off,
                                                     const float* sbase) {
  // GVS mode: global address is the 64-bit SGPR base plus the 32-bit VGPR offset; the VDST VGPR gives the LDS address
  asm volatile("global_load_async_to_lds_b128 %0, %1, %2"
               :: "v"(lds_byte_off), "v"(glb_byte_off), "s"(sbase)
               : "memory");
}
__device__ __forceinline__ void wait_async_le1() {
  asm volatile("s_wait_asynccnt 0x1" ::: "memory");
}
__device__ __forceinline__ void wait_async_0() {
  asm volatile("s_wait_asynccnt 0x0" ::: "memory");
}

struct Top3 {
  float p[3];
  int   i[3];
};

__device__ __forceinline__ void top3_init(Top3& t) {
  t.p[0] = t.p[1] = t.p[2] = -1.0f;
  t.i[0] = t.i[1] = t.i[2] = 0x7fffffff;
}

// Matches jax.lax.top_k ordering: higher prob first; ties -> lower index.
__device__ __forceinline__ void top3_insert(Top3& t, float p, int idx) {
  if (p > t.p[0] || (p == t.p[0] && idx < t.i[0])) {
    t.p[2] = t.p[1]; t.i[2] = t.i[1];
    t.p[1] = t.p[0]; t.i[1] = t.i[0];
    t.p[0] = p;      t.i[0] = idx;
  } else if (p > t.p[1] || (p == t.p[1] && idx < t.i[1])) {
    t.p[2] = t.p[1]; t.i[2] = t.i[1];
    t.p[1] = p;      t.i[1] = idx;
  } else if (p > t.p[2] || (p == t.p[2] && idx < t.i[2])) {
    t.p[2] = p;      t.i[2] = idx;
  }
}

__device__ __forceinline__ void accum(float p, int idx, float& ent, Top3& t) {
  if (p > 0.0f) ent += -p * logf(p);
  top3_insert(t, p, idx);
}

__global__ __launch_bounds__(THREADS)
void soft_masking_kernel(const int*   __restrict__ x_t,
                         const float* __restrict__ probs,
                         const float* __restrict__ embed,
                         const float* __restrict__ omega_s,
                         const float* __restrict__ omega_a,
                         const float* __restrict__ omega_b,
                         float*       __restrict__ out) {
  const int pos = blockIdx.x;        // 0 .. B*S-1
  const int tid = threadIdx.x;       // 0 .. 255

  const int token = x_t[pos];
  const float4* erow_tok = (const float4*)(embed + (size_t)token * H_);
  float4*       orow     = (float4*)(out + (size_t)pos * H_);

  if (token != MASK_ID_) {
    // ---------------- hot path: pure gather-copy of the embedding row ----------------
    orow[tid] = erow_tok[tid];       // b128 load + b128 store, coalesced
    return;
  }

  // ---------------- cold path: masked position (expected ~0.16 per launch) ----------------
  __shared__ float pipe[2][CHUNK];   // 32 KB double buffer for async streaming
  __shared__ float s_ent[THREADS];
  __shared__ float s_p[THREADS * 3];
  __shared__ int   s_i[THREADS * 3];
  __shared__ float s_w[3];
  __shared__ int   s_id[3];
  __shared__ float s_lam;

  const float* prow = probs + (size_t)pos * V_;
  const uint32_t lds_base =
      (uint32_t)(uintptr_t)(void*)&pipe[0][0];   // generic LDS ptr: low 32 bits = LDS offset

  float ent = 0.0f;
  Top3 t;
  top3_init(t);

  // Double-buffered async global->LDS pipeline over 12 full 16KB chunks.
  // Each thread consumes exactly the float4 its own wave deposited -> per-wave
  // s_wait_asynccnt is sufficient, no __syncthreads in the streaming loop.
  {
    // prologue
    async_load_f4_to_lds(lds_base + (uint32_t)tid * 16u,
                         (uint32_t)tid * 16u, prow);
    for (int c = 0; c < NFULL; ++c) {
      if (c + 1 < NFULL) {
        uint32_t buf = (uint32_t)((c + 1) & 1) * (CHUNK * 4u);
        async_load_f4_to_lds(lds_base + buf + (uint32_t)tid * 16u,
                             (uint32_t)(c + 1) * (CHUNK * 4u) + (uint32_t)tid * 16u,
                             prow);
        wait_async_le1();            // chunk c complete (async loads complete in order)
      } else {
        wait_async_0();
      }
      const float* bufp = &pipe[c & 1][tid * 4];
      float4 v = *(const float4*)bufp;
      int base = c * CHUNK + tid * 4;
      accum(v.x, base + 0, ent, t);
      accum(v.y, base + 1, ent, t);
      accum(v.z, base + 2, ent, t);
      accum(v.w, base + 3, ent, t);
    }
  }

  // Tail: 1105 remaining floats, guarded direct loads (avoids OOB of the allocation).
  for (int i = TAIL_START + tid; i < V_; i += THREADS) {
    accum(prow[i], i, ent, t);
  }

  // Publish partials.
  s_ent[tid] = ent;
  s_p[tid * 3 + 0] = t.p[0]; s_i[tid * 3 + 0] = t.i[0];
  s_p[tid * 3 + 1] = t.p[1]; s_i[tid * 3 + 1] = t.i[1];
  s_p[tid * 3 + 2] = t.p[2]; s_i[tid * 3 + 2] = t.i[2];
  __syncthreads();

  // Entropy tree reduction.
  for (int off = THREADS / 2; off > 0; off >>= 1) {
    if (tid < off) s_ent[tid] += s_ent[tid + off];
    __syncthreads();
  }

  // Thread 0: merge 768 top-3 candidates, compute weights and lambda.
  if (tid == 0) {
    Top3 g;
    top3_init(g);
    for (int e = 0; e < THREADS * 3; ++e) top3_insert(g, s_p[e], s_i[e]);

    float sum = g.p[0] + g.p[1] + g.p[2] + 1e-10f;
    s_w[0] = g.p[0] / sum;  s_id[0] = g.i[0];
    s_w[1] = g.p[1] / sum;  s_id[1] = g.i[1];
    s_w[2] = g.p[2] / sum;  s_id[2] = g.i[2];

    float entropy = s_ent[0];
    float os  = 1.0f / (1.0f + expf(-omega_s[0]));        // sigmoid(omega_s)
    float oa  = log1pf(expf(omega_a[0]));                 // softplus(omega_a)
    float obp = log1pf(expf(omega_b[0]));                 // softplus(omega_b)  (real_ob = -obp)
    // lam = os * sigmoid(oa * (-entropy - real_ob)) = os * sigmoid(oa * (obp - entropy))
    s_lam = os / (1.0f + expf(-(oa * (obp - entropy))));
  }
  __syncthreads();

  const float lam = s_lam;
  const float w0 = s_w[0], w1 = s_w[1], w2 = s_w[2];
  const float4 em = erow_tok[tid];   // token == MASK_ID here: real embedding of the mask token
  const float4 e0 = ((const float4*)(embed + (size_t)s_id[0] * H_))[tid];
  const float4 e1 = ((const float4*)(embed + (size_t)s_id[1] * H_))[tid];
  const float4 e2 = ((const float4*)(embed + (size_t)s_id[2] * H_))[tid];

  float4 r;
  r.x = (1.0f - lam) * em.x + lam * (w0 * e0.x + w1 * e1.x + w2 * e2.x);
  r.y = (1.0f - lam) * em.y + lam * (w0 * e0.y + w1 * e1.y + w2 * e2.y);
  r.z = (1.0f - lam) * em.z + lam * (w0 * e0.z + w1 * e1.z + w2 * e2.z);
  r.w = (1.0f - lam) * em.w + lam * (w0 * e0.w + w1 * e1.w + w2 * e2.w);
  orow[tid] = r;
}

extern "C" void kernel_launch(void* const* d_in, const int* in_sizes, int n_in,
                              void* d_out, int out_size, void* d_ws, size_t ws_size,
                              hipStream_t stream) {
  (void)in_sizes; (void)n_in; (void)out_size; (void)d_ws; (void)ws_size;
  const int*   x_t     = (const int*)d_in[0];
  const float* probs   = (const float*)d_in[1];
  const float* embed   = (const float*)d_in[2];
  const float* omega_s = (const float*)d_in[3];
  const float* omega_a = (const float*)d_in[4];
  const float* omega_b = (const float*)d_in[5];
  float* out = (float*)d_out;

  dim3 grid(B_ * S_);   // 8192 blocks, one per (b, s) position
  dim3 block(THREADS);  // 8 wave32s
  soft_masking_kernel<<<grid, block, 0, stream>>>(x_t, probs, embed,
                                                  omega_s, omega_a, omega_b, out);
}